// NeuralVoiceDecoder_4286377361807
// MI455X (gfx1250) — compile-verified
//
#include <hip/hip_runtime.h>
#include <math.h>

// ---------------- problem constants ----------------
constexpr int   NB     = 32;             // batch
constexpr int   FRAMES = 1000;
constexpr int   HOP    = 256;
constexpr int   L      = FRAMES * HOP;   // 256000
constexpr int   NFFT   = 1024;
constexpr int   BINS   = 513;            // rfft bins
constexpr int   NBINS_C= 65;             // curve bins
constexpr int   KPAD   = 528;            // 513 padded to multiple of 16
constexpr int   NFWD   = 2 * KPAD;       // 1056  (Re | Im)
constexpr int   NFWDP  = 1152;           // 1056 padded to multiple of 128 (9*128)
constexpr int   T      = 1001;           // stft frames
constexpr int   M      = NB * T;         // 32032 rows  (= 1001 * 32)
constexpr int   OUTLEN = NFFT + HOP * (T - 1); // 257024
constexpr int   PADH   = NFFT / 2;       // 512
constexpr float SRf    = 16000.0f;
constexpr float PI_F   = 3.14159265358979323846f;

typedef float v2f __attribute__((ext_vector_type(2)));
typedef float v8f __attribute__((ext_vector_type(8)));

static inline int cdiv(long long a, int b) { return (int)((a + b - 1) / b); }

// ---------------- basis generation ----------------
// Forward: Wf[j][c], j in [0,1024), c in [0,1152).
//   c <  528          : k=c,     val = (k<513) ?  cos(2*pi*j*k/1024) : 0
//   528 <= c < 1056   : k=c-528, val = (k<513) ? -sin(2*pi*j*k/1024) : 0
//   c >= 1056         : 0 (tile padding)
__global__ void k_basis_fwd(float* __restrict__ Wf) {
    int gid = blockIdx.x * blockDim.x + threadIdx.x;
    if (gid >= NFFT * NFWDP) return;
    int j = gid / NFWDP, c = gid % NFWDP;
    float v = 0.0f;
    if (c < KPAD) {
        int k = c;
        if (k < BINS) { int m = (j * k) & (NFFT - 1); v = cosf(2.0f * PI_F * (float)m / (float)NFFT); }
    } else if (c < NFWD) {
        int k = c - KPAD;
        if (k < BINS) { int m = (j * k) & (NFFT - 1); v = -sinf(2.0f * PI_F * (float)m / (float)NFFT); }
    }
    Wf[gid] = v;
}

// Inverse: Bi[r][n], r in [0,1056), n in [0,1024).
// y[n] = (1/N) * sum_k ck*(Re_k*cos(2pi k n/N) - Im_k*sin(2pi k n/N)), ck = 1 for k=0,512 else 2.
__global__ void k_basis_inv(float* __restrict__ Bi) {
    int gid = blockIdx.x * blockDim.x + threadIdx.x;
    if (gid >= NFWD * NFFT) return;
    int r = gid / NFFT, n = gid % NFFT;
    float v = 0.0f;
    if (r < KPAD) {
        int k = r;
        if (k < BINS) {
            float ck = (k == 0 || k == NFFT / 2) ? 1.0f : 2.0f;
            int m = (k * n) & (NFFT - 1);
            v = ck * cosf(2.0f * PI_F * (float)m / (float)NFFT) * (1.0f / (float)NFFT);
        }
    } else {
        int k = r - KPAD;
        if (k < BINS) {
            float ck = (k == 0 || k == NFFT / 2) ? 1.0f : 2.0f;
            int m = (k * n) & (NFFT - 1);
            v = -ck * sinf(2.0f * PI_F * (float)m / (float)NFFT) * (1.0f / (float)NFFT);
        }
    }
    Bi[gid] = v;
}

__device__ inline float hann_w(int j) {
    return 0.5f * (1.0f - cosf(2.0f * PI_F * (float)j / (float)NFFT));
}

// window[0..1023] and wsq[0..OUTLEN)
__global__ void k_win_wsq(float* __restrict__ win, float* __restrict__ wsq) {
    int n = blockIdx.x * blockDim.x + threadIdx.x;
    if (n >= OUTLEN) return;
    if (n < NFFT) win[n] = hann_w(n);
    int tlo = n - (NFFT - 1);
    int tmin = (tlo <= 0) ? 0 : (tlo + HOP - 1) / HOP;
    int tmax = n / HOP; if (tmax > T - 1) tmax = T - 1;
    float s = 0.0f;
    for (int t = tmin; t <= tmax; ++t) { float w = hann_w(n - t * HOP); s += w * w; }
    wsq[n] = s;
}

// ---------------- parameter upsampling (1000 -> 256000) ----------------
__global__ void k_upsample(const float* __restrict__ f0, const float* __restrict__ amp,
                           const float* __restrict__ oq,
                           float* __restrict__ f0u, float* __restrict__ ampu, float* __restrict__ oqu) {
    int gid = blockIdx.x * blockDim.x + threadIdx.x;
    if (gid >= NB * L) return;
    int b = gid / L, i = gid % L;
    float pos = ((float)i + 0.5f) * ((float)FRAMES / (float)L) - 0.5f;
    pos = fminf(fmaxf(pos, 0.0f), (float)(FRAMES - 1));
    int i0 = (int)floorf(pos);
    int i1 = min(i0 + 1, FRAMES - 1);
    float w = pos - (float)i0;
    int base = b * FRAMES;
    f0u[gid]  = f0[base + i0]  * (1.0f - w) + f0[base + i1]  * w;
    ampu[gid] = amp[base + i0] * (1.0f - w) + amp[base + i1] * w;
    oqu[gid]  = oq[base + i0]  * (1.0f - w) + oq[base + i1]  * w;
}

// ---------------- phase cumsum + Rosenberg pulse ----------------
// one block (256 threads = 8 wave32) per batch row; Hillis-Steele scan per 256-chunk
__global__ void k_glottal(const float* __restrict__ f0u, const float* __restrict__ oqu,
                          float* __restrict__ g) {
    __shared__ float sdata[256];
    int b = blockIdx.x;
    int tid = threadIdx.x;
    float carry = 0.0f;
    const float inv_sr = 1.0f / SRf;
    for (int chunk = 0; chunk < L / 256; ++chunk) {
        int i = chunk * 256 + tid;
        float x = f0u[b * L + i] * inv_sr;
        sdata[tid] = x; __syncthreads();
        #pragma unroll
        for (int off = 1; off < 256; off <<= 1) {
            float v = (tid >= off) ? sdata[tid - off] : 0.0f;
            __syncthreads();
            sdata[tid] += v;
            __syncthreads();
        }
        float phase = carry + sdata[tid];
        float tot = sdata[255];
        float p = phase - floorf(phase);
        float oqv = fminf(fmaxf(oqu[b * L + i], 0.1f), 0.9f);
        float pulse = 0.5f * (1.0f - cosf(PI_F * (p / (oqv + 1e-8f))));
        float mask = 1.0f / (1.0f + expf(-(oqv - p) * 100.0f));
        g[b * L + i] = pulse * mask;
        __syncthreads();
        carry += tot;
    }
}

// ---------------- framing (reflect pad, window, diff-on-the-fly) ----------------
__global__ void k_frame(const float* __restrict__ sigbuf, const float* __restrict__ win,
                        float* __restrict__ frames, int useGlottalDiff) {
    int gid = blockIdx.x * blockDim.x + threadIdx.x;
    if (gid >= M * NFFT) return;
    int row = gid >> 10;            // /1024
    int j   = gid & (NFFT - 1);
    int b = row / T, t = row % T;
    int pos = t * HOP + j - PADH;
    if (pos < 0) pos = -pos;
    else if (pos >= L) pos = 2 * L - 2 - pos;
    const float* s = sigbuf + (size_t)b * L;
    float v;
    if (useGlottalDiff) v = (pos == 0) ? 0.0f : (s[pos] - s[pos - 1]);
    else                v = s[pos];
    frames[gid] = v * win[j];
}

// ---------------- LDS-tiled WMMA f32 GEMM ----------------
// C[M x N] = A[M x K] * B[K x N], strides lda/ldb/ldc.
// Block: 128 threads = 4 wave32; macro-tile 32 (M) x 128 (N); each wave one
// 32x32 tile = 2x2 of 16x16 WMMA f32 subtiles. K chunk = 32, register-staged
// software pipeline (next chunk's global loads overlap current chunk's WMMAs).
// Requires: M%32==0, N%128==0, K%32==0 (guaranteed by padded problem dims).
constexpr int GBM = 32, GBN = 128, GKC = 32;

__global__ __launch_bounds__(128)
void k_gemm_tiled(const float* __restrict__ A, int lda,
                  const float* __restrict__ B, int ldb,
                  float* __restrict__ C, int ldc,
                  int tiles_n, int K) {
    __shared__ float As[GBM][36];    // 36-float row pitch: 16B-aligned, gcd(36,64)=4 -> conflict-free b64 reads
    __shared__ float Bs[GKC][132];   // 132-float row pitch: 16B-aligned, offsets rows across banks

    const int tid  = threadIdx.x;
    const int lane = tid & 31;
    const int wv   = tid >> 5;              // 0..3 : n-subtile of the macro tile
    const int mt   = blockIdx.x / tiles_n;
    const int nt   = blockIdx.x % tiles_n;  // n fastest -> A macro-rows stay hot in L2
    const int m0   = mt * GBM;
    const int n0b  = nt * GBN;

    const int mrow  = lane & 15;
    const int khalf = (lane >> 4) << 1;     // {0,2}
    const int ncol  = lane & 15;
    const int wn    = wv * 32;

    v8f acc00 = {}, acc01 = {}, acc10 = {}, acc11 = {};

    float4 ra[2], rb[8];                    // register staging

    const int nchunks = K / GKC;

    // prologue: fetch chunk 0 into registers
    #pragma unroll
    for (int i = 0; i < 2; ++i) {
        int f = tid + i * 128, r = f >> 3, c4 = (f & 7) << 2;
        ra[i] = *(const float4*)(A + (size_t)(m0 + r) * lda + c4);
    }
    #pragma unroll
    for (int i = 0; i < 8; ++i) {
        int f = tid + i * 128, kk = f >> 5, c4 = (f & 31) << 2;
        rb[i] = *(const float4*)(B + (size_t)kk * ldb + n0b + c4);
    }

    for (int kc = 0; kc < nchunks; ++kc) {
        __syncthreads();
        // commit staged chunk to LDS
        #pragma unroll
        for (int i = 0; i < 2; ++i) {
            int f = tid + i * 128, r = f >> 3, c4 = (f & 7) << 2;
            *(float4*)(&As[r][c4]) = ra[i];
        }
        #pragma unroll
        for (int i = 0; i < 8; ++i) {
            int f = tid + i * 128, kk = f >> 5, c4 = (f & 31) << 2;
            *(float4*)(&Bs[kk][c4]) = rb[i];
        }
        __syncthreads();

        // stage next chunk (overlaps the WMMA work below)
        if (kc + 1 < nchunks) {
            int k0 = (kc + 1) * GKC;
            #pragma unroll
            for (int i = 0; i < 2; ++i) {
                int f = tid + i * 128, r = f >> 3, c4 = (f & 7) << 2;
                ra[i] = *(const float4*)(A + (size_t)(m0 + r) * lda + k0 + c4);
            }
            #pragma unroll
            for (int i = 0; i < 8; ++i) {
                int f = tid + i * 128, kk = f >> 5, c4 = (f & 31) << 2;
                rb[i] = *(const float4*)(B + (size_t)(k0 + kk) * ldb + n0b + c4);
            }
        }

        const float* a0p = &As[mrow][khalf];
        const float* a1p = &As[16 + mrow][khalf];
        #pragma unroll
        for (int ks = 0; ks < GKC; ks += 4) {
            v2f a0 = *(const v2f*)(a0p + ks);            // ds_load_b64
            v2f a1 = *(const v2f*)(a1p + ks);
            v2f b0, b1;
            b0.x = Bs[ks + khalf][wn + ncol];
            b0.y = Bs[ks + khalf + 1][wn + ncol];
            b1.x = Bs[ks + khalf][wn + 16 + ncol];
            b1.y = Bs[ks + khalf + 1][wn + 16 + ncol];
            acc00 = __builtin_amdgcn_wmma_f32_16x16x4_f32(false, a0, false, b0, (short)0, acc00, false, false);
            acc01 = __builtin_amdgcn_wmma_f32_16x16x4_f32(false, a0, false, b1, (short)0, acc01, false, false);
            acc10 = __builtin_amdgcn_wmma_f32_16x16x4_f32(false, a1, false, b0, (short)0, acc10, false, false);
            acc11 = __builtin_amdgcn_wmma_f32_16x16x4_f32(false, a1, false, b1, (short)0, acc11, false, false);
        }
    }

    const int mb = (lane >> 4) << 3;        // {0,8}
    #pragma unroll
    for (int i = 0; i < 8; ++i) {
        size_t r0 = (size_t)(m0 + mb + i) * ldc + n0b + wn + ncol;
        size_t r1 = (size_t)(m0 + 16 + mb + i) * ldc + n0b + wn + ncol;
        C[r0]      = acc00[i];
        C[r0 + 16] = acc01[i];
        C[r1]      = acc10[i];
        C[r1 + 16] = acc11[i];
    }
}

// ---------------- spectral filter (separable bilinear 1000x65 -> 1001x513) ----------------
__global__ void k_filter(const float* __restrict__ curve, float* __restrict__ spec) {
    int gid = blockIdx.x * blockDim.x + threadIdx.x;
    if (gid >= M * BINS) return;
    int row = gid / BINS, k = gid % BINS;
    int b = row / T, t = row % T;
    float posT = ((float)t + 0.5f) * ((float)FRAMES / (float)T) - 0.5f;
    posT = fminf(fmaxf(posT, 0.0f), (float)(FRAMES - 1));
    int iT0 = (int)floorf(posT); int iT1 = min(iT0 + 1, FRAMES - 1);
    float wT = posT - (float)iT0;
    float posK = ((float)k + 0.5f) * ((float)NBINS_C / (float)BINS) - 0.5f;
    posK = fminf(fmaxf(posK, 0.0f), (float)(NBINS_C - 1));
    int iK0 = (int)floorf(posK); int iK1 = min(iK0 + 1, NBINS_C - 1);
    float wK = posK - (float)iK0;
    const float* cb = curve + (size_t)b * FRAMES * NBINS_C;
    float f00 = cb[iT0 * NBINS_C + iK0], f01 = cb[iT0 * NBINS_C + iK1];
    float f10 = cb[iT1 * NBINS_C + iK0], f11 = cb[iT1 * NBINS_C + iK1];
    float filt = (1.0f - wT) * ((1.0f - wK) * f00 + wK * f01)
               +         wT  * ((1.0f - wK) * f10 + wK * f11);
    size_t s = (size_t)row * NFWDP;
    spec[s + k]        *= filt;   // Re
    spec[s + KPAD + k] *= filt;   // Im
}

// ---------------- windowed overlap-add ----------------
__global__ void k_oadd(const float* __restrict__ tbuf, const float* __restrict__ win,
                       float* __restrict__ acc) {
    int gid = blockIdx.x * blockDim.x + threadIdx.x;
    if (gid >= M * NFFT) return;
    int row = gid >> 10;
    int j   = gid & (NFFT - 1);
    int b = row / T, t = row % T;
    atomicAdd(&acc[(size_t)b * OUTLEN + t * HOP + j], tbuf[gid] * win[j]);
}

__global__ void k_zero(float* __restrict__ p, long long n) {
    long long gid = (long long)blockIdx.x * blockDim.x + threadIdx.x;
    if (gid < n) p[gid] = 0.0f;
}

// ---------------- final combine ----------------
__global__ void k_final(const float* __restrict__ accV, const float* __restrict__ accU,
                        const float* __restrict__ wsq, const float* __restrict__ ampu,
                        float* __restrict__ out) {
    int gid = blockIdx.x * blockDim.x + threadIdx.x;
    if (gid >= NB * L) return;
    int b = gid / L, i = gid % L;
    size_t n = (size_t)b * OUTLEN + PADH + i;
    float ws = fmaxf(wsq[PADH + i], 1e-11f);
    out[gid] = (accV[n] + accU[n]) / ws * ampu[gid];
}

// ---------------- host orchestration ----------------
extern "C" void kernel_launch(void* const* d_in, const int* in_sizes, int n_in,
                              void* d_out, int out_size, void* d_ws, size_t ws_size,
                              hipStream_t stream) {
    const float* f0    = (const float*)d_in[0];
    const float* amp   = (const float*)d_in[1];
    const float* oq    = (const float*)d_in[2];
    const float* vtc   = (const float*)d_in[3];
    const float* nfc   = (const float*)d_in[4];
    const float* noise = (const float*)d_in[5];
    float* out = (float*)d_out;

    float* ws = (float*)d_ws;
    float* Wf     = ws; ws += (size_t)NFFT * NFWDP;   // 1024 x 1152 forward DFT basis
    float* Bi     = ws; ws += (size_t)NFWD * NFFT;    // 1056 x 1024 inverse DFT basis
    float* win    = ws; ws += NFFT;
    float* wsq    = ws; ws += OUTLEN;
    float* f0u    = ws; ws += (size_t)NB * L;
    float* ampu   = ws; ws += (size_t)NB * L;
    float* oqu    = ws; ws += (size_t)NB * L;
    float* g      = ws; ws += (size_t)NB * L;
    float* frames = ws; ws += (size_t)M * NFFT;       // reused as inverse-DFT output
    float* spec   = ws; ws += (size_t)M * NFWDP;      // stride 1152
    float* accV   = ws; ws += (size_t)NB * OUTLEN;
    float* accU   = ws; ws += (size_t)NB * OUTLEN;

    const int TPB = 256;

    k_basis_fwd<<<cdiv((long long)NFFT * NFWDP, TPB), TPB, 0, stream>>>(Wf);
    k_basis_inv<<<cdiv((long long)NFWD * NFFT, TPB), TPB, 0, stream>>>(Bi);
    k_win_wsq  <<<cdiv(OUTLEN, TPB), TPB, 0, stream>>>(win, wsq);
    k_upsample <<<cdiv((long long)NB * L, TPB), TPB, 0, stream>>>(f0, amp, oq, f0u, ampu, oqu);
    k_glottal  <<<NB, 256, 0, stream>>>(f0u, oqu, g);
    k_zero     <<<cdiv(2LL * NB * OUTLEN, TPB), TPB, 0, stream>>>(accV, 2LL * NB * OUTLEN);

    const long long fele = (long long)M * NFFT;
    const int tiles_m = M / GBM;          // 1001

    for (int sig = 0; sig < 2; ++sig) {
        const float* src   = sig == 0 ? g   : noise;
        const float* curve = sig == 0 ? vtc : nfc;
        float* acc         = sig == 0 ? accV : accU;

        k_frame<<<cdiv(fele, TPB), TPB, 0, stream>>>(src, win, frames, sig == 0 ? 1 : 0);

        { // forward DFT: [M x 1024] * [1024 x 1152] -> spec
            int tn = NFWDP / GBN;         // 9
            k_gemm_tiled<<<tiles_m * tn, 128, 0, stream>>>(
                frames, NFFT, Wf, NFWDP, spec, NFWDP, tn, NFFT);
        }

        k_filter<<<cdiv((long long)M * BINS, TPB), TPB, 0, stream>>>(curve, spec);

        { // inverse DFT: [M x 1056] * [1056 x 1024] -> frames (reuse)
            int tn = NFFT / GBN;          // 8
            k_gemm_tiled<<<tiles_m * tn, 128, 0, stream>>>(
                spec, NFWDP, Bi, NFFT, frames, NFFT, tn, NFWD);
        }

        k_oadd<<<cdiv(fele, TPB), TPB, 0, stream>>>(frames, win, acc);
    }

    k_final<<<cdiv((long long)NB * L, TPB), TPB, 0, stream>>>(accV, accU, wsq, ampu, out);
}